// model_5514738008446
// MI455X (gfx1250) — compile-verified
//
#include <hip/hip_runtime.h>
#include <hip/hip_bf16.h>

#define VOCAB 100000
#define DIM   300
#define BATCH 8192
#define NPOS  10      // 2*WINDOW
#define NNEG  50      // 2*WINDOW*NEG
#define NTOT  60      // rows gathered per batch element (padded to 64)

typedef __attribute__((ext_vector_type(2))) float v2f;
typedef __attribute__((ext_vector_type(8))) float v8f;

__device__ __forceinline__ float log_sigmoid(float x) {
    // numerically stable: min(x,0) - log1p(exp(-|x|))
    return fminf(x, 0.0f) - log1pf(__expf(-fabsf(x)));
}

// One block (128 threads = 4 waves) per batch element.
// Wave w computes scores for context rows 16w .. 16w+15 via f32 WMMA:
//   D(16x16) += A(16x4 rows-of-o_emb) * B(4x16 i_vec replicated across columns)
__global__ void __launch_bounds__(128)
sgns_loss_kernel(const float* __restrict__ i_emb,
                 const float* __restrict__ o_emb,
                 const int*   __restrict__ i_word,
                 const int*   __restrict__ o_word,
                 const int*   __restrict__ n_word,
                 float*       __restrict__ per_batch_loss)
{
    __shared__ __align__(16) float s_ivec[DIM + 4];
    __shared__ float s_part[8];

    const int b    = blockIdx.x;
    const int tid  = threadIdx.x;
    const int lane = tid & 31;
    const int w    = tid >> 5;

    // Stage center vector in LDS (1200 B).
    const float* iv = i_emb + (long long)i_word[b] * DIM;
    for (int t = tid; t < DIM; t += 128) s_ivec[t] = iv[t];
    __syncthreads();

    // Each lane pair (L, L+16) owns one context row of this wave's 16-row tile.
    const int m = lane & 15;
    const int g = 16 * w + m;                 // global row id within the 60 (64 padded)
    int word = 0;                             // rows 60..63 -> vocab row 0 (all zeros)
    if (g < NPOS)      word = o_word[b * NPOS + g];
    else if (g < NTOT) word = n_word[b * NNEG + (g - NPOS)];
    const float* rowp = o_emb + (long long)word * DIM;
    const int kk = (lane < 16) ? 0 : 2;       // A-matrix K split across lane halves

    v8f acc = {};
#pragma unroll 5
    for (int t = 0; t < DIM / 4; ++t) {
        const int k = 4 * t + kk;             // k is even -> 8B aligned
        const float2 av = *(const float2*)(rowp + k);       // global_load_b64 (row walk)
        const float2 bv = *(const float2*)(&s_ivec[k]);     // ds_load_b64
        v2f a;  a.x  = av.x; a.y  = av.y;
        v2f bb; bb.x = bv.x; bb.y = bv.y;     // B replicated: every column of D equal
        // 8 args: (neg_a, A, neg_b, B, c_mod, C, reuse_a, reuse_b)
        acc = __builtin_amdgcn_wmma_f32_16x16x4_f32(
            false, a, false, bb, (short)0, acc, false, false);
    }

    // D layout: VGPR r -> M=r on lanes 0-15, M=8+r on lanes 16-31 (all columns equal).
    float s = 0.0f;
    const int mb = (lane < 16) ? 0 : 8;
#pragma unroll
    for (int r = 0; r < 8; ++r) {
        const int   gr = 16 * w + mb + r;
        const float sc = acc[r];
        if (gr < NPOS)      s += log_sigmoid(sc);    // positive context
        else if (gr < NTOT) s += log_sigmoid(-sc);   // negative sample (row was negated)
    }
    if ((lane & 15) == 0) s_part[(w << 1) | (lane >> 4)] = s;
    __syncthreads();

    if (tid == 0) {
        float tot = 0.0f;
#pragma unroll
        for (int i = 0; i < 8; ++i) tot += s_part[i];
        per_batch_loss[b] = -0.1f * tot;   // -(oloss + nloss) for this b, both are (1/10)*sums
    }
}

// Deterministic single-block tree reduction -> batch mean.
__global__ void __launch_bounds__(256)
reduce_mean_kernel(const float* __restrict__ per_batch_loss, float* __restrict__ out)
{
    __shared__ float sh[256];
    float s = 0.0f;
    for (int i = threadIdx.x; i < BATCH; i += 256) s += per_batch_loss[i];
    sh[threadIdx.x] = s;
    __syncthreads();
    for (int off = 128; off > 0; off >>= 1) {
        if ((int)threadIdx.x < off) sh[threadIdx.x] += sh[threadIdx.x + off];
        __syncthreads();
    }
    if (threadIdx.x == 0) out[0] = sh[0] * (1.0f / BATCH);
}

extern "C" void kernel_launch(void* const* d_in, const int* in_sizes, int n_in,
                              void* d_out, int out_size, void* d_ws, size_t ws_size,
                              hipStream_t stream)
{
    const float* i_emb  = (const float*)d_in[0];
    const float* o_emb  = (const float*)d_in[1];
    const int*   i_word = (const int*)  d_in[2];
    const int*   o_word = (const int*)  d_in[3];
    const int*   n_word = (const int*)  d_in[4];
    float*       ws     = (float*)d_ws;          // BATCH floats of scratch

    sgns_loss_kernel<<<BATCH, 128, 0, stream>>>(i_emb, o_emb, i_word, o_word, n_word, ws);
    reduce_mean_kernel<<<1, 256, 0, stream>>>(ws, (float*)d_out);
}